// GraphPoolModule_78683800863046
// MI455X (gfx1250) — compile-verified
//
#include <hip/hip_runtime.h>

typedef __attribute__((ext_vector_type(2))) float v2f;
typedef __attribute__((ext_vector_type(8))) float v8f;

#define C_DIM 128  // feature channels (fixed by reference)

// One wave (32 lanes) per (16-segment group) x (64-channel tile).
// out[16 x 16] tiles accumulated via V_WMMA_F32_16X16X4_F32 (full fp32):
//   A[m,k] = (edge k belongs to segment m) ? 1/deg[m] : 0   (mean folded in)
//   B[k,n] = input[idxn[k], chan_n]
// seg_ids is sorted, so 16 consecutive segments own one contiguous edge range
// found with two binary searches (no prefix-scan pass needed).
__global__ __launch_bounds__(32)
void seg_mean_wmma_kernel(const float* __restrict__ input,
                          const int*   __restrict__ idxn,
                          const int*   __restrict__ seg_ids,
                          const int*   __restrict__ degs,
                          float*       __restrict__ out,
                          int n_out, int E)
{
    const int lane  = (int)threadIdx.x;      // 0..31
    const int m     = lane & 15;             // A row / B col within the 16-tile
    const int kb    = (lane >> 4) * 2;       // K pair handled by this lane: 0 or 2
    const int s0    = (int)blockIdx.x * 16;  // first segment of this group
    const int cbase = (int)blockIdx.y * 64;  // first channel of this tile
    const int myseg = s0 + m;

    // Per-lane weight for A rows (lane L serves row M = L&15).
    float w = 0.0f;
    if (myseg < n_out) {
        int d = degs[myseg];
        w = (d > 0) ? (1.0f / (float)d) : 0.0f;
    }

    // Binary-search the contiguous edge range of segments [s0, s0+16).
    // All-32-bit: E ~ 1.6M fits easily. Uniform across the wave.
    int eBeg, eEnd;
    {
        const int vLo = s0;
        const int vHi = (s0 + 16 < n_out) ? (s0 + 16) : n_out;
        int lo = 0, hi = E;
        while (lo < hi) { int mid = (lo + hi) >> 1; if (seg_ids[mid] < vLo) lo = mid + 1; else hi = mid; }
        eBeg = lo;
        hi = E;
        while (lo < hi) { int mid = (lo + hi) >> 1; if (seg_ids[mid] < vHi) lo = mid + 1; else hi = mid; }
        eEnd = lo;
    }

    v8f acc0 = {}, acc1 = {}, acc2 = {}, acc3 = {};

    const int eLast = eEnd - 1;              // safe clamp target (loop empty if eEnd==eBeg)

    for (int e = eBeg; e < eEnd; e += 8) {
        // Speculative prefetch of the index streams (global_prefetch_b8);
        // OOB prefetches are silently dropped, so no guard branch needed.
        __builtin_prefetch(seg_ids + e + 256, 0, 3);
        __builtin_prefetch(idxn    + e + 256, 0, 3);

        #pragma unroll
        for (int half = 0; half < 2; ++half) {
            const int e0  = e + half * 4 + kb;
            const int e1  = e0 + 1;
            // Branchless tail handling: clamp load index, zero the A weight.
            const int le0 = (e0 < eLast) ? e0 : eLast;
            const int le1 = (e1 < eLast) ? e1 : eLast;

            const int sg0 = seg_ids[le0];
            const int sg1 = seg_ids[le1];
            const int ix0 = idxn[le0];
            const int ix1 = idxn[le1];

            // A fragment: selection * (1/deg); zero annihilates garbage in B.
            v2f A;
            A.x = ((e0 < eEnd) & (sg0 == myseg)) ? w : 0.0f;
            A.y = ((e1 < eEnd) & (sg1 == myseg)) ? w : 0.0f;

            // B fragments: lane n reads channel (cbase + blk*16 + n) of the two
            // gathered rows; lanes 0-15 hit 16 consecutive dwords -> coalesced.
            const float* r0 = input + (ix0 << 7) + cbase + m;   // 32-bit voffset
            const float* r1 = input + (ix1 << 7) + cbase + m;
            v2f B0 = { r0[0],  r1[0]  };
            v2f B1 = { r0[16], r1[16] };
            v2f B2 = { r0[32], r1[32] };
            v2f B3 = { r0[48], r1[48] };

            acc0 = __builtin_amdgcn_wmma_f32_16x16x4_f32(false, A, false, B0, (short)0, acc0, false, false);
            acc1 = __builtin_amdgcn_wmma_f32_16x16x4_f32(false, A, false, B1, (short)0, acc1, false, false);
            acc2 = __builtin_amdgcn_wmma_f32_16x16x4_f32(false, A, false, B2, (short)0, acc2, false, false);
            acc3 = __builtin_amdgcn_wmma_f32_16x16x4_f32(false, A, false, B3, (short)0, acc3, false, false);
        }
    }

    // C/D layout: VGPR r -> row (r + 8*(lane>>4)), col = lane&15.
    const int rowHi = (lane >> 4) * 8;
    #pragma unroll
    for (int r = 0; r < 8; ++r) {
        const int s = s0 + r + rowHi;
        if (s < n_out) {
            float* o = out + s * C_DIM + cbase + m;
            o[0]  = acc0[r];
            o[16] = acc1[r];
            o[32] = acc2[r];
            o[48] = acc3[r];
        }
    }
}

extern "C" void kernel_launch(void* const* d_in, const int* in_sizes, int n_in,
                              void* d_out, int out_size, void* d_ws, size_t ws_size,
                              hipStream_t stream) {
    const float* input   = (const float*)d_in[0];
    const int*   idxn    = (const int*)  d_in[1];
    const int*   seg_ids = (const int*)  d_in[2];
    const int*   degs    = (const int*)  d_in[3];
    float*       out     = (float*)d_out;

    const int E     = in_sizes[1];
    const int n_out = in_sizes[3];

    dim3 grid((unsigned)((n_out + 15) / 16), C_DIM / 64, 1);
    dim3 block(32, 1, 1);
    seg_mean_wmma_kernel<<<grid, block, 0, stream>>>(input, idxn, seg_ids, degs,
                                                     out, n_out, E);
}